// LMEncoder_54778012893619
// MI455X (gfx1250) — compile-verified
//
#include <hip/hip_runtime.h>

typedef __attribute__((ext_vector_type(2))) float v2f;
typedef __attribute__((ext_vector_type(4))) float v4f;
typedef __attribute__((ext_vector_type(8))) float v8f;

#define LL 4            // layers
#define BB 32           // batch
#define SS 512          // sequence
#define DD 768          // hidden dim
#define TT 16           // output rows per tile (WMMA M)
#define KPAD 20         // gather rows per tile: 19 needed (t0+1 .. t0+19), padded to mult of 4
#define LDSS (DD + 8)   // 776: 2*776 % 64 == 16 -> half-wave LDS reads hit disjoint banks
#define NTHREADS 256    // 8 wave32 waves

__global__ __launch_bounds__(NTHREADS)
void lm_span_pool_wmma(const float* __restrict__ h,
                       const int* __restrict__ spans,
                       const int* __restrict__ masks,
                       float* __restrict__ out) {
  __shared__ float xtile[KPAD * LDSS];
  __shared__ int s_masklen;

  const int tid       = threadIdx.x;
  const int tilesPerB = SS / TT;
  const int b         = blockIdx.x / tilesPerB;
  const int t0        = (blockIdx.x % tilesPerB) * TT;
  const int t1        = t0 + 1;

  if (tid == 0) s_masklen = 0;
  __syncthreads();

  // ---- mask_len[b] = sum(masks[b, :]) ----
  {
    int part = 0;
    for (int i = tid; i < SS; i += NTHREADS) part += masks[b * SS + i];
    atomicAdd(&s_masklen, part);
  }

  // ---- Xtile[row][d] = mean_L hidden[l, b, t1+row, d] into LDS (zero beyond S) ----
  // hidden_states is streamed exactly once -> non-temporal b128 loads keep L2 clean.
  const int F4 = KPAD * (DD / 4);
  for (int f = tid; f < F4; f += NTHREADS) {
    const int row = f / (DD / 4);
    const int c4  = f - row * (DD / 4);
    const int r   = t1 + row;
    v4f acc = {0.f, 0.f, 0.f, 0.f};
    if (r < SS) {
#pragma unroll
      for (int l = 0; l < LL; ++l) {
        const v4f* p = (const v4f*)(h + (((size_t)l * BB + b) * SS + r) * DD) + c4;
        acc += __builtin_nontemporal_load(p);
      }
      acc *= 0.25f;
    }
    *(v4f*)&xtile[row * LDSS + c4 * 4] = acc;
  }
  __syncthreads();

  // ---- banded-matrix span pool via V_WMMA_F32_16X16X4_F32 ----
  const int wave  = tid >> 5;
  const int lane  = tid & 31;
  const int mlane = lane & 15;   // A row (M) / B,D column (N) owned by this lane
  const int hi    = lane >> 4;   // half-wave: K pair {0,1} vs {2,3}; D rows M vs M+8

  // Branch-free per-lane metadata (keeps EXEC uniform for the WMMA phase).
  const int t       = t0 + mlane;                 // output row for A
  const int sidx    = (t + 1 < SS) ? (t + 1) : (SS - 1);
  int       span    = spans[b * SS + sidx];
  const int masklen = s_masklen;
  // fold validity (t < masklen-2, span_next defined only for t+1 < S) into span
  if (t + 1 >= SS || t >= masklen - 2) span = 0;

  // A fragments are invariant across N-tiles: precompute all KPAD/4 of them.
  v2f af[KPAD / 4];
#pragma unroll
  for (int kc = 0; kc < KPAD / 4; ++kc) {
    const int k0 = kc * 4 + hi * 2;    // global K index of this lane's first element
    const int j0 = k0 - mlane;         // gather offset j = k - m
    af[kc].x = (j0 >= 0     && j0 < span && (t1 + k0)     < SS) ? 1.f : 0.f;
    af[kc].y = (j0 + 1 >= 0 && j0 + 1 < span && (t1 + k0 + 1) < SS) ? 1.f : 0.f;
  }

#pragma unroll
  for (int i = 0; i < 6; ++i) {          // 8 waves x 6 = 48 N-tiles = 768 cols
    const int n0 = wave * 96 + i * 16;
    v8f acc = {};
#pragma unroll
    for (int kc = 0; kc < KPAD / 4; ++kc) {
      const int k0 = kc * 4 + hi * 2;
      // B fragment (4x16 f32): lanes 0-15 rows K=k0,k0+1; lanes 16-31 rows K=k0+2,k0+3
      v2f bb;
      bb.x = xtile[k0       * LDSS + n0 + mlane];
      bb.y = xtile[(k0 + 1) * LDSS + n0 + mlane];
      acc = __builtin_amdgcn_wmma_f32_16x16x4_f32(false, af[kc], false, bb, (short)0,
                                                  acc, false, false);
    }
    // D layout: VGPR v -> M = v (lanes 0-15) / v+8 (lanes 16-31), N = mlane
#pragma unroll
    for (int v = 0; v < 8; ++v) {
      const int m = v + hi * 8;
      float* dst = out + ((size_t)b * SS + t0 + m) * DD + n0 + mlane;
      __builtin_nontemporal_store(acc[v], dst);
    }
  }
}

extern "C" void kernel_launch(void* const* d_in, const int* in_sizes, int n_in,
                              void* d_out, int out_size, void* d_ws, size_t ws_size,
                              hipStream_t stream) {
  (void)in_sizes; (void)n_in; (void)out_size; (void)d_ws; (void)ws_size;
  const float* h     = (const float*)d_in[0];
  const int*   spans = (const int*)d_in[1];
  const int*   masks = (const int*)d_in[2];
  float*       out   = (float*)d_out;

  dim3 grid(BB * (SS / TT));   // 1024 workgroups
  lm_span_pool_wmma<<<grid, NTHREADS, 0, stream>>>(h, spans, masks, out);
}